// LSNet2_75737453298096
// MI455X (gfx1250) — compile-verified
//
#include <hip/hip_runtime.h>
#include <hip/hip_bf16.h>
#include <math.h>

// ---------------------------------------------------------------------------
// LSNet2 morphsnakes step, MI455X (gfx1250, wave32).
// Fully fused morphology (6 passes + balloon dilation) in one LDS-resident
// kernel staged via CDNA5 async global->LDS loads; exact 40th percentile via
// deterministic integer radix select; Chan-Vese reductions + update.
// ---------------------------------------------------------------------------

#define OUT   64               // output tile edge
#define HALO  7                // 6 morph passes + 1 for the flat dilation
#define LW    (OUT + 2 * HALO) // 78: staged tile edge
#define PIT   80               // LDS row pitch (floats)

__device__ __forceinline__ float max6f(float a, float b, float c, float d, float e, float f) {
    return fmaxf(fmaxf(fmaxf(a, b), fmaxf(c, d)), fmaxf(e, f));
}
__device__ __forceinline__ float min6f(float a, float b, float c, float d, float e, float f) {
    return fminf(fminf(fminf(a, b), fminf(c, d)), fminf(e, f));
}

// One 3x3 morphological pass over the LDS tile.
// MODE: 0 = dilate_min (4 line SEs), 1 = erode_max (4 line SEs).
// O: inset of the region computed this pass (compile-time -> const div/mod).
template <int MODE, int O>
__device__ __forceinline__ void morph_pass(const float* sSrc, float* sDst,
                                           int gy0, int gx0, int H, int W, int tid) {
    constexpr int S = LW - 2 * O;
    for (int idx = tid; idx < S * S; idx += 256) {
        const int ty = O + idx / S;
        const int tx = O + idx % S;
        const float* c = &sSrc[ty * PIT + tx];
        const float n00 = c[-PIT - 1], n01 = c[-PIT], n02 = c[-PIT + 1];
        const float n10 = c[-1],       n11 = c[0],    n12 = c[1];
        const float n20 = c[ PIT - 1], n21 = c[ PIT], n22 = c[ PIT + 1];
        float val;
        if (MODE == 0) {
            // dilate per filter: max over 9 of (n + w); then min over the 4 filters
            const float d0 = fmaxf(fmaxf(fmaxf(n10, n11), n12) + 1.0f, max6f(n00, n01, n02, n20, n21, n22));
            const float d1 = fmaxf(fmaxf(fmaxf(n01, n11), n21) + 1.0f, max6f(n00, n02, n10, n12, n20, n22));
            const float d2 = fmaxf(fmaxf(fmaxf(n00, n11), n22) + 1.0f, max6f(n01, n02, n10, n12, n20, n21));
            const float d3 = fmaxf(fmaxf(fmaxf(n02, n11), n20) + 1.0f, max6f(n00, n01, n10, n12, n21, n22));
            val = fminf(fminf(d0, d1), fminf(d2, d3));
        } else {
            // erode per filter: min over 9 of (n - w); then max over the 4 filters
            const float e0 = fminf(fminf(fminf(n10, n11), n12) - 1.0f, min6f(n00, n01, n02, n20, n21, n22));
            const float e1 = fminf(fminf(fminf(n01, n11), n21) - 1.0f, min6f(n00, n02, n10, n12, n20, n22));
            const float e2 = fminf(fminf(fminf(n00, n11), n22) - 1.0f, min6f(n01, n02, n10, n12, n20, n21));
            const float e3 = fminf(fminf(fminf(n02, n11), n20) - 1.0f, min6f(n00, n01, n10, n12, n21, n22));
            val = fmaxf(fmaxf(e0, e1), fmaxf(e2, e3));
        }
        // positions outside the image are re-padded to 0 every pass (jnp.pad)
        const int gy = gy0 + ty, gx = gx0 + tx;
        if (!((unsigned)gy < (unsigned)H && (unsigned)gx < (unsigned)W)) val = 0.0f;
        sDst[ty * PIT + tx] = val;
    }
    __syncthreads();
}

// Fused: stage 78x78 halo tile (async global->LDS), run D,E,E,D,D,E in LDS,
// then flat 3x3 balloon dilation; emit x (64x64) and aux (64x64).
__global__ __launch_bounds__(256) void morph_fused(const float* __restrict__ src,
                                                   float* __restrict__ xOut,
                                                   float* __restrict__ auxOut,
                                                   int H, int W) {
    __shared__ float smA[LW * PIT];
    __shared__ float smB[LW * PIT];
    const int tid    = threadIdx.x;
    const int tilesX = W / OUT;
    const int bx     = blockIdx.x % tilesX;
    const int by     = blockIdx.x / tilesX;
    const int gx0    = bx * OUT - HALO;
    const int gy0    = by * OUT - HALO;

    // Stage the halo tile into LDS with CDNA5 async global->LDS loads.
    for (int idx = tid; idx < LW * LW; idx += 256) {
        const int ty = idx / LW;
        const int tx = idx - ty * LW;
        const int gy = gy0 + ty;
        const int gx = gx0 + tx;
        const unsigned ldsOff = (unsigned)(size_t)(&smA[ty * PIT + tx]);
        if ((unsigned)gy < (unsigned)H && (unsigned)gx < (unsigned)W) {
            const unsigned voff = (unsigned)(gy * W + gx) * 4u;
            // GVS mode: mem_addr = SGPR base + VGPR offset; LDS dest addr in VGPR.
            asm volatile("global_load_async_to_lds_b32 %0, %1, %2"
                         :: "v"(ldsOff), "v"(voff), "s"(src)
                         : "memory");
        } else {
            smA[ty * PIT + tx] = 0.0f;  // zero padding
        }
    }
    asm volatile("s_wait_asynccnt 0" ::: "memory");
    __syncthreads();

    // SI o IS smoothing: D, E, E, D, D, E (ping-pong smA <-> smB)
    morph_pass<0, 1>(smA, smB, gy0, gx0, H, W, tid);
    morph_pass<1, 2>(smB, smA, gy0, gx0, H, W, tid);
    morph_pass<1, 3>(smA, smB, gy0, gx0, H, W, tid);
    morph_pass<0, 4>(smB, smA, gy0, gx0, H, W, tid);
    morph_pass<0, 5>(smA, smB, gy0, gx0, H, W, tid);
    morph_pass<1, 6>(smB, smA, gy0, gx0, H, W, tid);
    // x now lives in smA over region [6,72)

    // balloon: aux = 1 + max over flat 3x3 of zero-padded x; also emit x
    const int col = tid & (OUT - 1);
    for (int r = tid >> 6; r < OUT; r += 4) {
        const float* c = &smA[(HALO + r) * PIT + (HALO + col)];
        const float xv  = c[0];
        const float aux = 1.0f + fmaxf(fmaxf(fmaxf(c[-PIT - 1], c[-PIT]), fmaxf(c[-PIT + 1], c[-1])),
                                       fmaxf(fmaxf(xv, c[1]), fmaxf(fmaxf(c[PIT - 1], c[PIT]), c[PIT + 1])));
        const int g = (by * OUT + r) * W + (bx * OUT + col);
        xOut[g]   = xv;
        auxOut[g] = aux;
    }
}

// ------------------------- exact percentile: radix select ------------------
__device__ __forceinline__ unsigned fkey(float f) {
    unsigned u = __float_as_uint(f);
    return (u & 0x80000000u) ? ~u : (u | 0x80000000u);
}
__device__ __forceinline__ float funkey(unsigned u) {
    unsigned v = (u & 0x80000000u) ? (u & 0x7fffffffu) : ~u;
    return __uint_as_float(v);
}

// ctrl layout (unsigned words): [0]=prefix [1]=mask [2]=remaining-rank [8..263]=bins
__global__ __launch_bounds__(256) void radix_init(unsigned* ctrl, unsigned rank) {
    ctrl[8 + threadIdx.x] = 0u;
    if (threadIdx.x == 0) { ctrl[0] = 0u; ctrl[1] = 0u; ctrl[2] = rank; }
}

__global__ __launch_bounds__(256) void radix_count(const float4* __restrict__ x4, int n4,
                                                   unsigned* __restrict__ ctrl, int shift) {
    __shared__ unsigned lb[256];
    lb[threadIdx.x] = 0u;
    __syncthreads();
    const unsigned prefix = ctrl[0];
    const unsigned mask   = ctrl[1];
    const int stride = gridDim.x * blockDim.x;
    for (int i = blockIdx.x * blockDim.x + threadIdx.x; i < n4; i += stride) {
        const float4 v = x4[i];
        unsigned u;
        u = fkey(v.x); if ((u & mask) == prefix) atomicAdd(&lb[(u >> shift) & 255u], 1u);
        u = fkey(v.y); if ((u & mask) == prefix) atomicAdd(&lb[(u >> shift) & 255u], 1u);
        u = fkey(v.z); if ((u & mask) == prefix) atomicAdd(&lb[(u >> shift) & 255u], 1u);
        u = fkey(v.w); if ((u & mask) == prefix) atomicAdd(&lb[(u >> shift) & 255u], 1u);
    }
    __syncthreads();
    const unsigned c = lb[threadIdx.x];
    if (c) atomicAdd(&ctrl[8 + threadIdx.x], c);
}

__global__ void radix_scan(unsigned* ctrl, int shift, float* out_val) {
    if (threadIdx.x == 0 && blockIdx.x == 0) {
        unsigned k = ctrl[2], cum = 0u;
        int b = 0;
        for (; b < 255; ++b) {
            const unsigned c = ctrl[8 + b];
            if (cum + c > k) break;
            cum += c;
        }
        ctrl[2]  = k - cum;
        ctrl[0] |= ((unsigned)b) << shift;
        ctrl[1] |= 0xFFu << shift;
        for (int i = 0; i < 256; ++i) ctrl[8 + i] = 0u;
        if (out_val) *out_val = funkey(ctrl[0]);
    }
}

// F layout (floats): [0]=v_k [1]=v_{k+1} [2]=threshold [3]=c0 [4]=c1
__global__ void thresh_kernel(float* F, float frac) {
    if (threadIdx.x == 0 && blockIdx.x == 0) F[2] = F[0] + frac * (F[1] - F[0]);
}

// ------------- fused balloon-select + Chan-Vese partial sums ---------------
__global__ __launch_bounds__(256) void fuse_where_sums(const float* __restrict__ x,
                                                       const float* __restrict__ aux,
                                                       const float* __restrict__ origin,
                                                       const float* __restrict__ F,
                                                       float* __restrict__ x1,
                                                       float* __restrict__ partials,
                                                       int n) {
    __shared__ float s0[256], s1[256], s2[256];
    const float thr = F[2];
    float a0 = 0.0f, a1 = 0.0f, a2 = 0.0f;
    const int stride = gridDim.x * blockDim.x;
    for (int i = blockIdx.x * blockDim.x + threadIdx.x; i < n; i += stride) {
        const float o  = origin[i];
        const float xv = x[i];
        const float v  = (o > thr) ? aux[i] : xv;
        x1[i] = v;
        a0 += o;
        a1 += o * v;
        a2 += v;
    }
    const int t = threadIdx.x;
    s0[t] = a0; s1[t] = a1; s2[t] = a2;
    __syncthreads();
    for (int s = 128; s > 0; s >>= 1) {
        if (t < s) { s0[t] += s0[t + s]; s1[t] += s1[t + s]; s2[t] += s2[t + s]; }
        __syncthreads();
    }
    if (t == 0) {
        const int G = gridDim.x;
        partials[blockIdx.x]         = s0[0];
        partials[G + blockIdx.x]     = s1[0];
        partials[2 * G + blockIdx.x] = s2[0];
    }
}

__global__ __launch_bounds__(256) void cv_means(const float* __restrict__ partials, int nb,
                                                float* __restrict__ F, float N) {
    __shared__ float s0[256], s1[256], s2[256];
    float a0 = 0.0f, a1 = 0.0f, a2 = 0.0f;
    for (int i = threadIdx.x; i < nb; i += 256) {
        a0 += partials[i];
        a1 += partials[nb + i];
        a2 += partials[2 * nb + i];
    }
    const int t = threadIdx.x;
    s0[t] = a0; s1[t] = a1; s2[t] = a2;
    __syncthreads();
    for (int s = 128; s > 0; s >>= 1) {
        if (t < s) { s0[t] += s0[t + s]; s1[t] += s1[t + s]; s2[t] += s2[t + s]; }
        __syncthreads();
    }
    if (t == 0) {
        const float sum_o   = s0[0];
        const float sum_ox1 = s1[0];
        const float sum_x1  = s2[0];
        F[3] = (sum_o - sum_ox1) / ((N - sum_x1) + 1e-8f);  // c0
        F[4] = sum_ox1 / (sum_x1 + 1e-8f);                  // c1
    }
}

__global__ __launch_bounds__(256) void cv_update(const float* __restrict__ origin,
                                                 const float* __restrict__ x1,
                                                 const float* __restrict__ F,
                                                 float* __restrict__ out, int n) {
    const float c0 = F[3];
    const float c1 = F[4];
    const int stride = gridDim.x * blockDim.x;
    for (int i = blockIdx.x * blockDim.x + threadIdx.x; i < n; i += stride) {
        const float o  = origin[i];
        const float d1 = o - c1;
        const float d0 = o - c0;
        const float cv = d1 * d1 - d0 * d0;
        out[i] = (cv < 0.0f) ? 1.0f : ((cv > 0.0f) ? 0.0f : x1[i]);
    }
}

// ---------------------------------------------------------------------------
extern "C" void kernel_launch(void* const* d_in, const int* in_sizes, int n_in,
                              void* d_out, int out_size, void* d_ws, size_t ws_size,
                              hipStream_t stream) {
    const float* input  = (const float*)d_in[0];
    const float* origin = (const float*)d_in[1];
    // d_in[2] = filters: structuring elements are fixed; hardcoded in kernels.

    const int n = in_sizes[0];
    const int W = (int)lround(sqrt((double)n));
    const int H = n / W;
    float* out = (float*)d_out;

    // workspace carve-up
    const size_t nBytes = (size_t)n * sizeof(float);
    char*     w        = (char*)d_ws;
    float*    B1       = (float*)(w);                       // aux
    unsigned* ctrl     = (unsigned*)(w + nBytes);           // radix-select control (264 u32)
    float*    F        = (float*)(w + nBytes + 2048);       // scalars
    float*    partials = (float*)(w + nBytes + 4096);       // 3 * RED_BLOCKS floats
    float*    B0       = (float*)(w + nBytes + 65536);      // x1 buffer

    const int tiles      = (H / OUT) * (W / OUT);
    const int RED_BLOCKS = 2048;
    const int SEL_BLOCKS = 1024;

    // percentile ranks + interpolation fraction (host-side, deterministic)
    const double qidx = 0.40 * (double)(n - 1);
    const unsigned kRank = (unsigned)floor(qidx);
    const float frac = (float)(qidx - (double)kRank);

    // ---- fused D,E,E,D,D,E + balloon dilation: x -> d_out, aux -> B1
    morph_fused<<<tiles, 256, 0, stream>>>(input, out, B1, H, W);

    // ---- exact 40th percentile of x: two order statistics via radix select
    radix_init<<<1, 256, 0, stream>>>(ctrl, kRank);
    for (int p = 0; p < 4; ++p) {
        const int shift = 24 - 8 * p;
        radix_count<<<SEL_BLOCKS, 256, 0, stream>>>((const float4*)out, n / 4, ctrl, shift);
        radix_scan<<<1, 1, 0, stream>>>(ctrl, shift, (shift == 0) ? &F[0] : (float*)nullptr);
    }
    radix_init<<<1, 256, 0, stream>>>(ctrl, kRank + 1u);
    for (int p = 0; p < 4; ++p) {
        const int shift = 24 - 8 * p;
        radix_count<<<SEL_BLOCKS, 256, 0, stream>>>((const float4*)out, n / 4, ctrl, shift);
        radix_scan<<<1, 1, 0, stream>>>(ctrl, shift, (shift == 0) ? &F[1] : (float*)nullptr);
    }
    thresh_kernel<<<1, 1, 0, stream>>>(F, frac);

    // ---- balloon select + Chan-Vese sums
    fuse_where_sums<<<RED_BLOCKS, 256, 0, stream>>>(out, B1, origin, F, B0, partials, n);
    cv_means<<<1, 256, 0, stream>>>(partials, RED_BLOCKS, F, (float)n);
    cv_update<<<RED_BLOCKS, 256, 0, stream>>>(origin, B0, F, out, n);

    (void)n_in; (void)out_size; (void)ws_size;
}